// SpectralClassifier_4475355922831
// MI455X (gfx1250) — compile-verified
//
#include <hip/hip_runtime.h>
#include <math.h>

typedef __attribute__((ext_vector_type(2))) float v2f;
typedef __attribute__((ext_vector_type(8))) float v8f;

#define BB 1024
#define SS 128
#define NN 127       // DCT length
#define DD 768
#define FILT_J 15    // keep k in [0, 15)

// ---------------------------------------------------------------------------
// Kernel 1: collapse DCT -> bandpass -> IDCT -> mean into a single 127-vector
// w[n] = (1/(127*254)) * sum_{k<15} colsum_M3[k] * C2[k][n]
// ---------------------------------------------------------------------------
__global__ void spectral_w_kernel(float* __restrict__ w) {
    __shared__ float colsum[FILT_J];
    const int t = threadIdx.x;
    const float PI = 3.14159265358979323846f;
    if (t < FILT_J) {
        float s;
        if (t == 0) {
            s = (float)NN;                       // M3[:,0] == 1
        } else {
            s = 0.0f;
            for (int m = 0; m < NN; ++m)
                s += 2.0f * cosf(PI * (float)t * (2.0f * m + 1.0f) / (2.0f * NN));
        }
        colsum[t] = s;
    }
    __syncthreads();
    if (t < NN) {
        float acc = 0.0f;
        for (int k = 0; k < FILT_J; ++k) {
            float c2 = 2.0f * cosf(PI * (float)k * (2.0f * t + 1.0f) / (2.0f * NN));
            acc += colsum[k] * c2;
        }
        w[t] = acc / ((float)NN * 2.0f * (float)NN);   // / (127 * 254)
    }
}

// ---------------------------------------------------------------------------
// Kernel 2: bandwidth-bound masked weighted pooling over the sequence axis.
// v[b,d] = sum_n w[n] * (mask[b,n+1] ? hs[b,n+1,d] : 0)
// grid (B, 3), block 256 -> d = by*256 + tid. Coalesced 1KB row segments.
// ---------------------------------------------------------------------------
__global__ void pool_kernel(const float* __restrict__ hs,
                            const int* __restrict__ mask,
                            const float* __restrict__ w,
                            float* __restrict__ v) {
    const int b = blockIdx.x;
    const int d = blockIdx.y * 256 + threadIdx.x;

    __shared__ float wm[NN];
    if (threadIdx.x < NN)
        wm[threadIdx.x] = mask[b * SS + 1 + threadIdx.x] ? w[threadIdx.x] : 0.0f;
    __syncthreads();

    const float* base = hs + (size_t)b * SS * DD + DD + d;  // row n=0 -> seq pos 1
    float acc = 0.0f;
#pragma unroll 8
    for (int n = 0; n < NN; ++n) {
        if (n + 16 < NN) __builtin_prefetch(base + (size_t)(n + 16) * DD, 0, 1);
        acc += wm[n] * base[(size_t)n * DD];
    }
    v[(size_t)b * DD + d] = acc;
}

// ---------------------------------------------------------------------------
// Kernel 3: f32 WMMA GEMM  C[M,N] = act(A[M,K] @ B[K,N] + bias)
// One wave32 owns one 16x16 output tile; K steps of 4 via v_wmma_f32_16x16x4_f32.
// A frag (16x4):  lane l (m = l&15, h = l>>4): a.x = A[m, k+2h], a.y = A[m, k+2h+1]
// B frag (4x16):  lane l (n = l&15, h = l>>4): b.x = B[k+2h, n], b.y = B[k+2h+1, n]
// C/D  (16x16):   VGPR r, lane l: element [r + 8h, n]
// ---------------------------------------------------------------------------
__global__ void gemm_wmma_f32(const float* __restrict__ A,
                              const float* __restrict__ B,
                              const float* __restrict__ bias,
                              float* __restrict__ C,
                              int M, int N, int K, int do_relu) {
    const int wave  = (blockIdx.x * blockDim.x + threadIdx.x) >> 5;
    const int lane  = threadIdx.x & 31;
    const int tilesN = N >> 4;
    const int totalTiles = (M >> 4) * tilesN;
    if (wave >= totalTiles) return;

    const int tm = wave / tilesN;
    const int tn = wave % tilesN;
    const int rc  = lane & 15;       // row within tile for A, col within tile for B/C/D
    const int h   = lane >> 4;
    const int row = tm * 16 + rc;
    const int col = tn * 16 + rc;

    v8f acc = {0.f, 0.f, 0.f, 0.f, 0.f, 0.f, 0.f, 0.f};

    for (int k = 0; k < K; k += 4) {
        // contiguous pair -> single 8B load
        v2f a = *(const v2f*)(A + (size_t)row * K + k + 2 * h);
        v2f b;
        b.x = B[(size_t)(k + 2 * h + 0) * N + col];
        b.y = B[(size_t)(k + 2 * h + 1) * N + col];
        acc = __builtin_amdgcn_wmma_f32_16x16x4_f32(
            /*neg_a=*/false, a, /*neg_b=*/false, b,
            /*c_mod=*/(short)0, acc, /*reuse_a=*/false, /*reuse_b=*/false);
    }

    const float bv = bias[col];
#pragma unroll
    for (int r = 0; r < 8; ++r) {
        float val = acc[r] + bv;
        if (do_relu) val = fmaxf(val, 0.0f);
        C[(size_t)(tm * 16 + r + 8 * h) * N + col] = val;
    }
}

// ---------------------------------------------------------------------------
// Kernel 4: tiny head  out[b, j] = h2[b, :] @ W3[:, j] + b3[j]   (64 -> 2)
// ---------------------------------------------------------------------------
__global__ void head_kernel(const float* __restrict__ h2,
                            const float* __restrict__ W3,
                            const float* __restrict__ b3,
                            float* __restrict__ out) {
    const int idx = blockIdx.x * 256 + threadIdx.x;
    if (idx >= BB * 2) return;
    const int b = idx >> 1;
    const int j = idx & 1;
    float acc = b3[j];
    const float* row = h2 + (size_t)b * 64;
#pragma unroll
    for (int k = 0; k < 64; ++k) acc += row[k] * W3[k * 2 + j];
    out[idx] = acc;
}

// ---------------------------------------------------------------------------
extern "C" void kernel_launch(void* const* d_in, const int* in_sizes, int n_in,
                              void* d_out, int out_size, void* d_ws, size_t ws_size,
                              hipStream_t stream) {
    const float* hs   = (const float*)d_in[0];   // [B, S, D]
    const int*   mask = (const int*)d_in[1];     // [B, S] bool -> int
    const float* W1   = (const float*)d_in[2];   // [768, 256]
    const float* b1   = (const float*)d_in[3];   // [256]
    const float* W2   = (const float*)d_in[4];   // [256, 64]
    const float* b2   = (const float*)d_in[5];   // [64]
    const float* W3   = (const float*)d_in[6];   // [64, 2]
    const float* b3   = (const float*)d_in[7];   // [2]
    float* out = (float*)d_out;                  // [B, 2]

    // workspace layout (512B-aligned slices)
    char* ws = (char*)d_ws;
    float* w_vec = (float*)(ws);                                  // 128 f32
    float* v     = (float*)(ws + 512);                            // 1024*768 f32
    float* h1    = (float*)(ws + 512 + (size_t)BB * DD * 4);      // 1024*256 f32
    float* h2    = (float*)(ws + 512 + (size_t)BB * DD * 4 + (size_t)BB * 256 * 4); // 1024*64

    // 1) spectral weights
    spectral_w_kernel<<<1, 128, 0, stream>>>(w_vec);

    // 2) masked weighted pooling (HBM-bound, ~400MB read)
    pool_kernel<<<dim3(BB, 3), 256, 0, stream>>>(hs, mask, w_vec, v);

    // 3) MLP layer 1: [1024,768] @ [768,256], relu.  (1024/16)*(256/16)=1024 waves
    {
        int waves = (BB / 16) * (256 / 16);
        gemm_wmma_f32<<<(waves * 32 + 255) / 256, 256, 0, stream>>>(
            v, W1, b1, h1, BB, 256, DD, 1);
    }

    // 4) MLP layer 2: [1024,256] @ [256,64], relu.  256 waves
    {
        int waves = (BB / 16) * (64 / 16);
        gemm_wmma_f32<<<(waves * 32 + 255) / 256, 256, 0, stream>>>(
            h1, W2, b2, h2, BB, 64, 256, 1);
    }

    // 5) head: [1024,64] @ [64,2]
    head_kernel<<<(BB * 2 + 255) / 256, 256, 0, stream>>>(h2, W3, b3, out);
}